// SECOND_BEAM_SEARCH_22333829940005
// MI455X (gfx1250) — compile-verified
//
#include <hip/hip_runtime.h>
#include <hip/hip_bf16.h>
#include <stdint.h>

// ---------------- problem constants ----------------
#define VOCAB   128000
#define BEAM    5
#define TOPK    5
#define HIST    511
#define NUM_KV  16
#define SLAB_FLOATS (8*2048*128)        // per-beam per-kv slab = 2,097,152 floats (8 MB)
#define SLAB_F4     (SLAB_FLOATS/4)     // 524,288 float4

// ---------------- output layout (flat floats, reference return order) ---------
#define OUT_KV_TOTAL ((long long)NUM_KV * BEAM * SLAB_FLOATS)     // 167,772,160
#define OFF_SAVE (OUT_KV_TOTAL)                                   // new_save [5,512]
#define OFF_RP   (OFF_SAVE + (long long)BEAM*512)                 // rp [5,128000]
#define OFF_PROB (OFF_RP + (long long)BEAM*VOCAB)                 // top_beam_prob [5,1]
#define OFF_TOK  (OFF_PROB + BEAM)                                // token_i [5,1]
#define OFF_MAX  (OFF_TOK + BEAM)                                 // max_logits_idx [1]

// ---------------- workspace layout ----------------
// ws+0   : float tkP[25]   (beam-major: b*5+j)
// ws+128 : int   tkI[25]
// ws+256 : int   beam_index[5]
// ws+288 : int   token[5]

// =====================================================================
// K1: penalized log-softmax normalizer + per-beam top-5 (one block/beam)
// =====================================================================
#define K1_T 512

__global__ __launch_bounds__(K1_T)
void k1_softmax_topk(const float* __restrict__ logits, const float* __restrict__ rp,
                     float* __restrict__ tkP, int* __restrict__ tkI)
{
    __shared__ float lm[K1_T], ls[K1_T];
    __shared__ float lv[K1_T][TOPK];
    __shared__ int   li[K1_T][TOPK];

    const int b = blockIdx.x, t = threadIdx.x;
    const float* lg = logits + (size_t)b * VOCAB;
    const float* rr = rp     + (size_t)b * VOCAB;

    float m = -__builtin_inff(), s = 0.f;
    float v5[TOPK]; int i5[TOPK];
#pragma unroll
    for (int j = 0; j < TOPK; ++j) { v5[j] = -__builtin_inff(); i5[j] = 0x7fffffff; }

    for (int v = t; v < VOCAB; v += K1_T) {
        float x = lg[v] * rr[v];
        // online max + sum(exp(x - max))
        if (x > m) { s = s * __expf(m - x) + 1.f; m = x; }
        else       { s += __expf(x - m); }
        // sorted-descending top-5 insert (strict > keeps lowest index on ties)
        if (x > v5[TOPK-1]) {
            int p = TOPK - 1;
            while (p > 0 && x > v5[p-1]) { v5[p] = v5[p-1]; i5[p] = i5[p-1]; --p; }
            v5[p] = x; i5[p] = v;
        }
    }

    lm[t] = m; ls[t] = s;
#pragma unroll
    for (int j = 0; j < TOPK; ++j) { lv[t][j] = v5[j]; li[t][j] = i5[j]; }
    __syncthreads();

    for (int str = K1_T / 2; str > 0; str >>= 1) {
        if (t < str) {
            float m2 = lm[t+str], s2 = ls[t+str];
            float M  = fmaxf(lm[t], m2);
            ls[t] = ls[t] * __expf(lm[t] - M) + s2 * __expf(m2 - M);
            lm[t] = M;
            float av[TOPK], bv[TOPK], rv[TOPK];
            int   ai[TOPK], bi2[TOPK], ri[TOPK];
#pragma unroll
            for (int j = 0; j < TOPK; ++j) {
                av[j] = lv[t][j];     ai[j]  = li[t][j];
                bv[j] = lv[t+str][j]; bi2[j] = li[t+str][j];
            }
            int pa = 0, pb = 0;
#pragma unroll
            for (int k = 0; k < TOPK; ++k) {
                bool ta = (av[pa] > bv[pb]) || (av[pa] == bv[pb] && ai[pa] < bi2[pb]);
                if (ta) { rv[k] = av[pa]; ri[k] = ai[pa]; ++pa; }
                else    { rv[k] = bv[pb]; ri[k] = bi2[pb]; ++pb; }
            }
#pragma unroll
            for (int j = 0; j < TOPK; ++j) { lv[t][j] = rv[j]; li[t][j] = ri[j]; }
        }
        __syncthreads();
    }

    if (t == 0) {
        float logZ = lm[0] + __logf(ls[0]);
#pragma unroll
        for (int j = 0; j < TOPK; ++j) {
            tkP[b*TOPK + j] = lv[0][j] - logZ;
            tkI[b*TOPK + j] = li[0][j];
        }
    }
}

// =====================================================================
// K2: combine 25 candidates -> 5 beams; small outputs + new_save gather
// =====================================================================
__global__ __launch_bounds__(256)
void k2_combine(const float* __restrict__ tkP, const int* __restrict__ tkI,
                const float* __restrict__ prevProb, const int* __restrict__ save_id,
                int* __restrict__ ws_bi, int* __restrict__ ws_tok,
                float* __restrict__ out_prob, float* __restrict__ out_tok,
                float* __restrict__ out_maxidx, float* __restrict__ out_save)
{
    __shared__ int s_bi[BEAM], s_tok[BEAM];

    if (threadIdx.x == 0) {
        float cur[BEAM*TOPK];
        bool  used[BEAM*TOPK];
#pragma unroll
        for (int i = 0; i < BEAM*TOPK; ++i) { cur[i] = tkP[i] + prevProb[i / TOPK]; used[i] = false; }
        for (int j = 0; j < BEAM; ++j) {
            int best = 0; float bv = -__builtin_inff();
            for (int i = 0; i < BEAM*TOPK; ++i)
                if (!used[i] && cur[i] > bv) { bv = cur[i]; best = i; }  // strict >: lowest idx on tie
            used[best] = true;
            int bi  = best / TOPK;
            int tok = tkI[best];
            s_bi[j] = bi;  s_tok[j] = tok;
            ws_bi[j] = bi; ws_tok[j] = tok;
            out_prob[j] = bv;
            out_tok[j]  = (float)tok;
            if (j == 0) out_maxidx[0] = (float)tok;
        }
    }
    __syncthreads();

    // new_save = concat(save_id[beam_index], token)  -> [5, 512]
    for (int idx = threadIdx.x; idx < BEAM * 512; idx += blockDim.x) {
        int b = idx >> 9, c = idx & 511;
        float val = (c < HIST) ? (float)save_id[s_bi[b]*HIST + c] : (float)s_tok[b];
        out_save[idx] = val;
    }
}

// =====================================================================
// K3: repeat_penality gather + penalty multiply at chosen token
// =====================================================================
__global__ __launch_bounds__(256)
void k3_rp(const float* __restrict__ rp, const int* __restrict__ ws_bi,
           const int* __restrict__ ws_tok, const float* __restrict__ pen,
           float* __restrict__ out_rp)
{
    long long i = (long long)blockIdx.x * blockDim.x + threadIdx.x;   // exact-size grid
    int b = (int)(i / VOCAB);
    int v = (int)(i % VOCAB);
    float x = rp[(size_t)ws_bi[b] * VOCAB + v];
    if (v == ws_tok[b]) x *= pen[0];
    out_rp[i] = x;
}

// =====================================================================
// K4: 1.34 GB KV beam-gather via CDNA5 async global<->LDS DMA.
//     80 slabs x 32 blocks; per thread 64 float4 in batches of 8
//     (one wave-wide async op = 32 lanes x 16 B = 512 B; a batch of 8
//      keeps 4 KB per wave in flight per ASYNCcnt fence).
//     Note: ASYNCcnt counts loads+stores and they complete out-of-order
//     relative to each other, so partial-count waits are ambiguous —
//     batches must be separated by `s_wait_asynccnt 0`. We amortize the
//     fence with bigger batches + ~20K resident waves.
// =====================================================================
#define BLKS_PER_SLAB 32
#define K4_T          256
#define K4_UNROLL     8
#define K4_ITERS      ((SLAB_F4 / BLKS_PER_SLAB) / (K4_T * K4_UNROLL))   // 8

__global__ __launch_bounds__(K4_T)
void k4_kv_gather(const float4* k0,  const float4* k1,  const float4* k2,  const float4* k3,
                  const float4* k4,  const float4* k5,  const float4* k6,  const float4* k7,
                  const float4* k8,  const float4* k9,  const float4* k10, const float4* k11,
                  const float4* k12, const float4* k13, const float4* k14, const float4* k15,
                  const int* __restrict__ ws_bi, float4* __restrict__ out)
{
    const int slab = blockIdx.x / BLKS_PER_SLAB;      // 0..79 : (kv, new_beam)
    const int part = blockIdx.x % BLKS_PER_SLAB;
    const int kv   = slab / BEAM;
    const int nb   = slab % BEAM;
    const int sb   = ws_bi[nb];                       // source beam (uniform per block)

    const float4* src;
    switch (kv) {
        case 0:  src = k0;  break; case 1:  src = k1;  break;
        case 2:  src = k2;  break; case 3:  src = k3;  break;
        case 4:  src = k4;  break; case 5:  src = k5;  break;
        case 6:  src = k6;  break; case 7:  src = k7;  break;
        case 8:  src = k8;  break; case 9:  src = k9;  break;
        case 10: src = k10; break; case 11: src = k11; break;
        case 12: src = k12; break; case 13: src = k13; break;
        default: src = (kv == 14) ? k14 : k15; break;
    }
    src += (size_t)sb * SLAB_F4;
    float4* dst = out + (size_t)slab * SLAB_F4;

    const unsigned base   = (unsigned)part * (SLAB_F4 / BLKS_PER_SLAB) + threadIdx.x;
    const unsigned lane16 = threadIdx.x * 16u;        // this thread's LDS byte slot

    for (int it = 0; it < K4_ITERS; ++it) {
#pragma unroll
        for (int u = 0; u < K4_UNROLL; ++u) {
            unsigned idx  = base + (unsigned)(it * K4_UNROLL + u) * K4_T;
            unsigned loff = lane16 + (unsigned)u * (K4_T * 16);
            uint64_t ga   = (uint64_t)(uintptr_t)(src + idx);
            asm volatile("global_load_async_to_lds_b128 %0, %1, off"
                         :: "v"(loff), "v"(ga) : "memory");
        }
        asm volatile("s_wait_asynccnt 0" ::: "memory");   // LDS now holds the tile
#pragma unroll
        for (int u = 0; u < K4_UNROLL; ++u) {
            unsigned idx  = base + (unsigned)(it * K4_UNROLL + u) * K4_T;
            unsigned loff = lane16 + (unsigned)u * (K4_T * 16);
            uint64_t ga   = (uint64_t)(uintptr_t)(dst + idx);
            asm volatile("global_store_async_from_lds_b128 %0, %1, off"
                         :: "v"(ga), "v"(loff) : "memory");
        }
        asm volatile("s_wait_asynccnt 0" ::: "memory");   // before reusing LDS slots
    }
}

// =====================================================================
extern "C" void kernel_launch(void* const* d_in, const int* in_sizes, int n_in,
                              void* d_out, int out_size, void* d_ws, size_t ws_size,
                              hipStream_t stream)
{
    (void)in_sizes; (void)n_in; (void)out_size; (void)ws_size;

    const float* kv[NUM_KV];
    for (int i = 0; i < NUM_KV; ++i) kv[i] = (const float*)d_in[i];
    const int*   save_id = (const int*)d_in[16];
    const float* rp      = (const float*)d_in[17];
    const float* prev    = (const float*)d_in[18];
    // d_in[19] = batch_indices (arange, unused)
    const float* logits  = (const float*)d_in[20];
    const float* pen     = (const float*)d_in[21];
    // d_in[22], d_in[23] = static beam_size/topK (compile-time constants here)

    float* out = (float*)d_out;
    char*  ws  = (char*)d_ws;
    float* tkP  = (float*)(ws + 0);
    int*   tkI  = (int*)(ws + 128);
    int*   wbi  = (int*)(ws + 256);
    int*   wtok = (int*)(ws + 288);

    k1_softmax_topk<<<BEAM, K1_T, 0, stream>>>(logits, rp, tkP, tkI);

    k2_combine<<<1, 256, 0, stream>>>(tkP, tkI, prev, save_id, wbi, wtok,
                                      out + OFF_PROB, out + OFF_TOK,
                                      out + OFF_MAX,  out + OFF_SAVE);

    // big gather first (dominant, ~58 us at 23.3 TB/s), 32 KB dynamic LDS per block
    k4_kv_gather<<<NUM_KV * BEAM * BLKS_PER_SLAB, K4_T, K4_T * 16 * K4_UNROLL, stream>>>(
        (const float4*)kv[0],  (const float4*)kv[1],  (const float4*)kv[2],  (const float4*)kv[3],
        (const float4*)kv[4],  (const float4*)kv[5],  (const float4*)kv[6],  (const float4*)kv[7],
        (const float4*)kv[8],  (const float4*)kv[9],  (const float4*)kv[10], (const float4*)kv[11],
        (const float4*)kv[12], (const float4*)kv[13], (const float4*)kv[14], (const float4*)kv[15],
        wbi, (float4*)out);

    k3_rp<<<(BEAM * VOCAB) / 256, 256, 0, stream>>>(rp, wbi, wtok, pen, out + OFF_RP);
}